// GNNEncoder_2637109919787
// MI455X (gfx1250) — compile-verified
//
#include <hip/hip_runtime.h>

#define NN 50000
#define NE 600000
#define D 128

typedef __attribute__((ext_vector_type(2))) float v2f;
typedef __attribute__((ext_vector_type(4))) float v4f;
typedef __attribute__((ext_vector_type(8))) float v8f;

// ---------------- utility kernels ----------------

__global__ void k_zero(v4f* __restrict__ p, int n4) {
    int i = blockIdx.x * blockDim.x + threadIdx.x;
    if (i < n4) { v4f z = {0.f, 0.f, 0.f, 0.f}; p[i] = z; }
}

__global__ void k_deg(const long long* __restrict__ dst, float* __restrict__ deg, int E) {
    int e = blockIdx.x * blockDim.x + threadIdx.x;
    if (e < E) unsafeAtomicAdd(&deg[(int)dst[e]], 1.0f);
}

__global__ void k_rdeg(float* __restrict__ deg, int n) {
    int i = blockIdx.x * blockDim.x + threadIdx.x;
    if (i < n) deg[i] = 1.0f / fmaxf(deg[i], 1.0f);
}

// 32 lanes per edge, each lane moves one float4 chunk (coalesced 512B per edge)
__global__ void k_scatter(const float* __restrict__ x, const long long* __restrict__ src,
                          const long long* __restrict__ dst, float* __restrict__ summed, int E) {
    int t = blockIdx.x * blockDim.x + threadIdx.x;
    int e = t >> 5;
    if (e >= E) return;
    int c = (t & 31) << 2;
    int s  = (int)src[e];
    int d0 = (int)dst[e];
    v4f v = *(const v4f*)(x + (size_t)s * D + c);
    float* o = summed + (size_t)d0 * D + c;
    unsafeAtomicAdd(o + 0, v.x);
    unsafeAtomicAdd(o + 1, v.y);
    unsafeAtomicAdd(o + 2, v.z);
    unsafeAtomicAdd(o + 3, v.w);
}

// Permute W[o][k] (row-major DxD) into the V_WMMA_F32_16X16X4_F32 B-fragment order:
// Bperm[(k>>1)*256 + o*2 + (k&1)] = W[o*D + k]
// so a lane's 2-element B fragment (K=k0,k0+1 at column n) is one contiguous v2f.
__global__ void k_wperm(const float* __restrict__ W, float* __restrict__ Bp) {
    int t = blockIdx.x * blockDim.x + threadIdx.x;
    if (t >= D * D) return;
    int o = t >> 7;
    int k = t & 127;
    Bp[((k >> 1) << 8) + (o << 1) + (k & 1)] = W[t];
}

// ---------------- fused SAGE layer GEMM ----------------
// out[i][:] = relu( (summed[i]*rdeg[i]) @ Wl^T + bl + xin[i] @ Wr^T )
// 8 waves/block; wave w owns rows [blk*128 + 16w, +16), all 128 output cols.
__global__ __launch_bounds__(256) void k_sage_gemm(
    const float* __restrict__ summed, const float* __restrict__ rdeg,
    const float* __restrict__ xin, const float* __restrict__ BpL,
    const float* __restrict__ BpR, const float* __restrict__ bias,
    float* __restrict__ out)
{
    __shared__ __align__(16) float Blds[D * D];   // 64KB: permuted weights for current pass

    const int tid  = threadIdx.x;
    const int lane = tid & 31;
    const int wave = tid >> 5;
    const int l16  = lane & 15;
    const int hi   = lane >> 4;                   // 0: K pair {0,1}; 1: K pair {2,3}
    const int rowbase = blockIdx.x * 128 + wave * 16;
    const int arow = min(rowbase + l16, NN - 1);  // clamped load row (EXEC stays all-1s)

    v8f acc[8];
#pragma unroll
    for (int nt = 0; nt < 8; ++nt) {
        float b = bias[nt * 16 + l16];            // bias depends only on output column
        v8f t = {b, b, b, b, b, b, b, b};
        acc[nt] = t;
    }

    v2f afrag[32];
#pragma unroll 1
    for (int pass = 0; pass < 2; ++pass) {
        // stage permuted weights into LDS (coalesced float4 copy)
        const float* Bsrc = pass ? BpR : BpL;
        for (int i = tid; i < D * D / 4; i += 256)
            ((v4f*)Blds)[i] = ((const v4f*)Bsrc)[i];
        __syncthreads();

        // A fragments for all 32 K-steps of this pass, scaled by 1/deg for the mean pass
        const float* A    = pass ? xin : summed;
        const float scale = pass ? 1.0f : rdeg[arow];
        const float* ap   = A + (size_t)arow * D + hi * 2;
#pragma unroll
        for (int kt = 0; kt < 32; ++kt) {
            v2f a = *(const v2f*)(ap + kt * 4);
            afrag[kt] = a * scale;
        }

#pragma unroll
        for (int nt = 0; nt < 8; ++nt) {
            const float* bp = Blds + ((nt * 16 + l16) << 1) + (hi << 8);
#pragma unroll
            for (int kt = 0; kt < 32; ++kt) {
                v2f b = *(const v2f*)(bp + (kt << 9));
                acc[nt] = __builtin_amdgcn_wmma_f32_16x16x4_f32(
                    false, afrag[kt], false, b, (short)0, acc[nt], false, false);
            }
        }
        __syncthreads();   // before next pass overwrites Blds
    }

    // epilogue: relu + store (C layout: VGPR r -> row r + 8*hi, col = nt*16 + l16)
#pragma unroll
    for (int nt = 0; nt < 8; ++nt) {
#pragma unroll
        for (int r = 0; r < 8; ++r) {
            int row = rowbase + hi * 8 + r;
            if (row < NN)
                out[(size_t)row * D + nt * 16 + l16] = fmaxf(acc[nt][r], 0.0f);
        }
    }
}

// ---------------- launch ----------------

extern "C" void kernel_launch(void* const* d_in, const int* in_sizes, int n_in,
                              void* d_out, int out_size, void* d_ws, size_t ws_size,
                              hipStream_t stream) {
    const float*     x  = (const float*)d_in[0];
    const long long* ei = (const long long*)d_in[1];
    const float* Wl[3] = {(const float*)d_in[2], (const float*)d_in[5], (const float*)d_in[8]};
    const float* bl[3] = {(const float*)d_in[3], (const float*)d_in[6], (const float*)d_in[9]};
    const float* Wr[3] = {(const float*)d_in[4], (const float*)d_in[7], (const float*)d_in[10]};

    const long long* src = ei;
    const long long* dst = ei + NE;

    float* rdeg   = (float*)d_ws;                         // NN (padded to 50048)
    float* summed = rdeg + 50048;                         // NN*D
    float* hA     = summed + (size_t)NN * D;              // NN*D
    float* hB     = hA + (size_t)NN * D;                  // NN*D
    float* BpL    = hB + (size_t)NN * D;                  // D*D
    float* BpR    = BpL + D * D;                          // D*D

    // 1/max(deg,1), computed once (edge list identical for all layers)
    k_zero<<<(NN / 4 + 255) / 256, 256, 0, stream>>>((v4f*)rdeg, NN / 4);
    k_deg<<<(NE + 255) / 256, 256, 0, stream>>>(dst, rdeg, NE);
    k_rdeg<<<(NN + 255) / 256, 256, 0, stream>>>(rdeg, NN);

    const float* cur = x;
    float* outs[3] = {hA, hB, (float*)d_out};
    for (int L = 0; L < 3; ++L) {
        k_zero<<<((size_t)NN * D / 4 + 255) / 256, 256, 0, stream>>>((v4f*)summed, NN * D / 4);
        k_scatter<<<((size_t)NE * 32 + 255) / 256, 256, 0, stream>>>(cur, src, dst, summed, NE);
        k_wperm<<<(D * D) / 256, 256, 0, stream>>>(Wl[L], BpL);
        k_wperm<<<(D * D) / 256, 256, 0, stream>>>(Wr[L], BpR);
        k_sage_gemm<<<(NN + 127) / 128, 256, 0, stream>>>(summed, rdeg, cur, BpL, BpR, bl[L], outs[L]);
        cur = outs[L];
    }
}